// FSGraphConv_47974784696630
// MI455X (gfx1250) — compile-verified
//
#include <hip/hip_runtime.h>
#include <hip/hip_bf16.h>
#include <math.h>

// GATv2 x2 + MLP head for MI455X (gfx1250, wave32).
// - node-sized state (<=45MB) stays L2-resident (192MB L2); edge streams use
//   non-temporal b128 loads.
// - dense GEMMs use fp32 WMMA (V_WMMA_F32_16X16X4_F32, exact f32 math),
//   B operands staged in LDS as paired float2 -> one ds_load_b64 per operand,
//   branch-free inner loops (no exec-mask thrash).

typedef float v2f __attribute__((ext_vector_type(2)));
typedef float v4f __attribute__((ext_vector_type(4)));
typedef float v8f __attribute__((ext_vector_type(8)));

#define NEG_SLOPE 0.2f

__device__ __forceinline__ v8f wmma4(v2f a, v2f b, v8f c) {
  // D = A(16x4 f32) * B(4x16 f32) + C(16x16 f32)
  return __builtin_amdgcn_wmma_f32_16x16x4_f32(
      false, a, false, b, (short)0, c, false, false);
}

__device__ __forceinline__ int imin(int a, int b) { return a < b ? a : b; }

__device__ __forceinline__ void atomic_max_f32(float* addr, float v) {
  if (v >= 0.0f) atomicMax((int*)addr, __float_as_int(v));
  else           atomicMin((unsigned int*)addr, __float_as_uint(v));
}

// ---------------------------------------------------------------- self loops
__global__ void k_loop_accum(const long long* __restrict__ dst,
                             const float* __restrict__ eattr,
                             float* __restrict__ cnt, float* __restrict__ lattr,
                             int E) {
  int e = blockIdx.x * blockDim.x + threadIdx.x;
  if (e >= E) return;
  long long d = __builtin_nontemporal_load(&dst[e]);
  atomicAdd(&cnt[d], 1.0f);
  const v4f* f4 = (const v4f*)(eattr + (size_t)e * 16);
  float* o = lattr + (size_t)d * 16;
#pragma unroll
  for (int i = 0; i < 4; i++) {
    v4f v = __builtin_nontemporal_load(&f4[i]);
#pragma unroll
    for (int j = 0; j < 4; j++) atomicAdd(&o[i * 4 + j], v[j]);
  }
}

__global__ void k_loop_div(float* __restrict__ lattr, const float* __restrict__ cnt,
                           int N) {
  int i = blockIdx.x * blockDim.x + threadIdx.x;
  if (i >= N * 16) return;
  lattr[i] /= fmaxf(cnt[i / 16], 1.0f);
}

// ---------------------------------------------------------------- init helpers
__global__ void k_init_bias(float* __restrict__ out, const float* __restrict__ bias,
                            int N, int stride, int Fout) {
  int i = blockIdx.x * blockDim.x + threadIdx.x;
  if (i >= N * stride) return;
  int j = i % stride;
  out[i] = (j < Fout) ? bias[j] : 0.0f;
}

__global__ void k_init_neginf(float* __restrict__ p, int n) {
  int i = blockIdx.x * blockDim.x + threadIdx.x;
  if (i < n) p[i] = -__builtin_huge_valf();
}

__global__ void k_relu(float* __restrict__ p, int n) {
  int i = blockIdx.x * blockDim.x + threadIdx.x;
  if (i < n) p[i] = fmaxf(p[i], 0.0f);
}

// ------------------------------------------------- layer-1 node transform (WMMA)
// xl/xr[N][16] = x[N][Fn] @ W{l,r}^T + b  (Fout=9 padded to 16; pad cols = 0)
// B staged in LDS as float2 pairs: sW[k/2][col] = {W^T[k][col], W^T[k+1][col]}
__global__ void k_node_lin1(const float* __restrict__ x,
                            const float* __restrict__ Wl, const float* __restrict__ bl,
                            const float* __restrict__ Wr, const float* __restrict__ br,
                            float* __restrict__ xl, float* __restrict__ xr,
                            int N, int Fn, int Fout) {
  __shared__ v2f sL[512];  // up to Fn=64 -> 32 pairs x 16 cols
  __shared__ v2f sR[512];
  int pairs = (Fn >> 1) * 16;
  for (int idx = threadIdx.x; idx < pairs; idx += blockDim.x) {
    int kk = idx >> 4, n = idx & 15;
    float m = (n < Fout) ? 1.0f : 0.0f;
    int nc = imin(n, Fout - 1);
    v2f l; l.x = Wl[nc * Fn + 2 * kk] * m; l.y = Wl[nc * Fn + 2 * kk + 1] * m;
    v2f r; r.x = Wr[nc * Fn + 2 * kk] * m; r.y = Wr[nc * Fn + 2 * kk + 1] * m;
    sL[idx] = l; sR[idx] = r;
  }
  __syncthreads();

  int wid = threadIdx.x >> 5, lane = threadIdx.x & 31;
  int tile = blockIdx.x * (blockDim.x >> 5) + wid;
  int r0 = tile * 16;
  if (r0 < N) {                        // wave-uniform
    int row = lane & 15;
    int koff = (lane < 16) ? 0 : 2;
    int col = lane & 15;
    int growc = imin(r0 + row, N - 1); // clamp: garbage rows never stored
    const float* xrow = x + (size_t)growc * Fn;
    v8f cl = {}; v8f cr = {};
    for (int k0 = 0; k0 < Fn; k0 += 4) {
      int k = k0 + koff;               // even -> 8B-aligned
      v2f a = *(const v2f*)(xrow + k);
      v2f bL = sL[(k >> 1) * 16 + col];
      v2f bR = sR[(k >> 1) * 16 + col];
      cl = wmma4(a, bL, cl);
      cr = wmma4(a, bR, cr);
    }
    bool cok = col < Fout;
    int rbase = (lane < 16) ? 0 : 8;
#pragma unroll
    for (int r = 0; r < 8; r++) {
      int gr = r0 + rbase + r;
      if (gr < N) {
        xl[(size_t)gr * 16 + col] = cok ? (cl[r] + bl[col]) : 0.0f;
        xr[(size_t)gr * 16 + col] = cok ? (cr[r] + br[col]) : 0.0f;
      }
    }
  }
}

// ------------------------------------------------- layer-2 node transform (WMMA)
// xl/xr[N][20] = h1[N][16](9 valid, zero-padded) @ W{l,r}(20x9)^T + b
__global__ void k_node_lin2(const float* __restrict__ h,
                            const float* __restrict__ Wl, const float* __restrict__ bl,
                            const float* __restrict__ Wr, const float* __restrict__ br,
                            float* __restrict__ xl, float* __restrict__ xr, int N) {
  const int Fin = 9, Fout = 20;
  __shared__ v2f sL[6 * 32];           // K padded to 12 -> 6 pairs x 32 cols
  __shared__ v2f sR[6 * 32];
  for (int idx = threadIdx.x; idx < 6 * 32; idx += blockDim.x) {
    int kk = idx >> 5, n = idx & 31;
    int k = 2 * kk;
    int nc = imin(n, Fout - 1);
    float m0 = (n < Fout && k < Fin) ? 1.0f : 0.0f;
    float m1 = (n < Fout && k + 1 < Fin) ? 1.0f : 0.0f;
    int k0c = imin(k, Fin - 1), k1c = imin(k + 1, Fin - 1);
    v2f l; l.x = Wl[nc * Fin + k0c] * m0; l.y = Wl[nc * Fin + k1c] * m1;
    v2f r; r.x = Wr[nc * Fin + k0c] * m0; r.y = Wr[nc * Fin + k1c] * m1;
    sL[idx] = l; sR[idx] = r;
  }
  __syncthreads();

  int wid = threadIdx.x >> 5, lane = threadIdx.x & 31;
  int tile = blockIdx.x * (blockDim.x >> 5) + wid;
  int r0 = tile * 16;
  if (r0 < N) {                        // wave-uniform
    int row = lane & 15;
    int koff = (lane < 16) ? 0 : 2;
    int c0 = lane & 15;
    int c1 = 16 + (lane & 15);
    int growc = imin(r0 + row, N - 1);
    const float* hrow = h + (size_t)growc * 16;
    v8f cl0 = {}, cl1 = {}, cr0 = {}, cr1 = {};
#pragma unroll
    for (int k0 = 0; k0 < 12; k0 += 4) {
      int k = k0 + koff;
      v2f a = *(const v2f*)(hrow + k); // h cols 9..15 are zero padding
      int kk = k >> 1;
      cl0 = wmma4(a, sL[kk * 32 + c0], cl0);
      cl1 = wmma4(a, sL[kk * 32 + c1], cl1);
      cr0 = wmma4(a, sR[kk * 32 + c0], cr0);
      cr1 = wmma4(a, sR[kk * 32 + c1], cr1);
    }
    bool c1ok = c1 < Fout;
    int rbase = (lane < 16) ? 0 : 8;
#pragma unroll
    for (int r = 0; r < 8; r++) {
      int gr = r0 + rbase + r;
      if (gr >= N) continue;
      xl[(size_t)gr * 20 + c0] = cl0[r] + bl[c0];
      xr[(size_t)gr * 20 + c0] = cr0[r] + br[c0];
      if (c1ok) {
        xl[(size_t)gr * 20 + c1] = cl1[r] + bl[c1];
        xr[(size_t)gr * 20 + c1] = cr1[r] + br[c1];
      }
    }
  }
}

// ---------------------------------------------------------------- edge pass 1
template <int H, int C, int SL>
__global__ void k_edge_logits(const long long* __restrict__ src,
                              const long long* __restrict__ dst,
                              const float* __restrict__ eattr,
                              const float* __restrict__ lattr,
                              const float* __restrict__ We,   // [H*C][16]
                              const float* __restrict__ att,  // [H][C]
                              const float* __restrict__ xl,
                              const float* __restrict__ xr,
                              float* __restrict__ logits,     // [H][EP]
                              float* __restrict__ lmax,       // [N][H]
                              int E, int EP) {
  constexpr int J = H * C;
  constexpr int NV = (J + 3) / 4;
  int e = blockIdx.x * blockDim.x + threadIdx.x;
  if (e >= EP) return;
  long long s, d; const float* fp;
  if (e < E) {
    s = __builtin_nontemporal_load(&src[e]);
    d = __builtin_nontemporal_load(&dst[e]);
    fp = eattr + (size_t)e * 16;
  } else {
    s = d = e - E;
    fp = lattr + (size_t)s * 16;
  }
  v4f f4[4];
#pragma unroll
  for (int i = 0; i < 4; i++) f4[i] = __builtin_nontemporal_load(&((const v4f*)fp)[i]);

  // L2-resident gathers; SL*4 bytes row stride is 16B-aligned for SL=16,20
  v4f xlv[NV], xrv[NV];
  const v4f* xls = (const v4f*)(xl + (size_t)s * SL);
  const v4f* xrd = (const v4f*)(xr + (size_t)d * SL);
#pragma unroll
  for (int i = 0; i < NV; i++) { xlv[i] = xls[i]; xrv[i] = xrd[i]; }

#pragma unroll
  for (int h = 0; h < H; h++) {
    float acc = 0.f;
#pragma unroll
    for (int c = 0; c < C; c++) {
      int j = h * C + c;
      float ea = 0.f;
      const float* wr = We + j * 16;
#pragma unroll
      for (int t = 0; t < 16; t++) ea += wr[t] * f4[t >> 2][t & 3];
      float m = xlv[j >> 2][j & 3] + xrv[j >> 2][j & 3] + ea;
      m = (m >= 0.f) ? m : NEG_SLOPE * m;
      acc += att[j] * m;
    }
    __builtin_nontemporal_store(acc, &logits[(size_t)h * EP + e]);
    atomic_max_f32(&lmax[(size_t)d * H + h], acc);
  }
}

// ---------------------------------------------------------------- edge pass 2
template <int H>
__global__ void k_edge_exp(const long long* __restrict__ dst,
                           float* __restrict__ logits, const float* __restrict__ lmax,
                           float* __restrict__ denom, int E, int EP) {
  int e = blockIdx.x * blockDim.x + threadIdx.x;
  if (e >= EP) return;
  long long d = (e < E) ? __builtin_nontemporal_load(&dst[e]) : (long long)(e - E);
#pragma unroll
  for (int h = 0; h < H; h++) {
    float v = __builtin_nontemporal_load(&logits[(size_t)h * EP + e]);
    float ex = __expf(v - lmax[(size_t)d * H + h]);
    __builtin_nontemporal_store(ex, &logits[(size_t)h * EP + e]);
    atomicAdd(&denom[(size_t)d * H + h], ex);
  }
}

// ---------------------------------------------------------------- edge pass 3
template <int H, int C, int SL, int SO>
__global__ void k_edge_scatter(const long long* __restrict__ src,
                               const long long* __restrict__ dst,
                               const float* __restrict__ logits,   // holds exp now
                               const float* __restrict__ denom,
                               const float* __restrict__ xl,
                               float* __restrict__ out,            // pre-init with bias
                               int E, int EP) {
  constexpr int J = H * C;
  constexpr int NV = (J + 3) / 4;
  int e = blockIdx.x * blockDim.x + threadIdx.x;
  if (e >= EP) return;
  long long s, d;
  if (e < E) {
    s = __builtin_nontemporal_load(&src[e]);
    d = __builtin_nontemporal_load(&dst[e]);
  } else {
    s = d = e - E;
  }
  v4f xlv[NV];
  const v4f* xls = (const v4f*)(xl + (size_t)s * SL);
#pragma unroll
  for (int i = 0; i < NV; i++) xlv[i] = xls[i];
#pragma unroll
  for (int h = 0; h < H; h++) {
    float alpha = __builtin_nontemporal_load(&logits[(size_t)h * EP + e]) /
                  denom[(size_t)d * H + h];
#pragma unroll
    for (int c = 0; c < C; c++) {
      int j = h * C + c;
      atomicAdd(&out[(size_t)d * SO + j], alpha * xlv[j >> 2][j & 3]);
    }
  }
}

// ---------------------------------------------------------------- FC1 (WMMA)
// acc[16][112] += A[16][K] @ W[F1][K]^T ; M=16 == batch, K split across waves
__global__ void k_fc1_wmma(const float* __restrict__ A, const float* __restrict__ W,
                           float* __restrict__ acc, int Bn, int K, int F1, int slices) {
  int wid = threadIdx.x >> 5, lane = threadIdx.x & 31;
  int nt = blockIdx.x;
  int slice = blockIdx.y * (blockDim.x >> 5) + wid;
  if (slice >= slices) return;         // wave-uniform
  int ksteps = K / 4;
  int per = (ksteps + slices - 1) / slices;
  int kbeg = slice * per;
  int kend = imin(kbeg + per, ksteps);
  int row = lane & 15;
  int koff = (lane < 16) ? 0 : 2;
  int col = nt * 16 + (lane & 15);
  int rowc = imin(row, Bn - 1);        // clamp; garbage rows ignored downstream
  int colc = imin(col, F1 - 1);
  float msk = (col < F1) ? 1.0f : 0.0f;
  const float* Ar = A + (size_t)rowc * K;
  const float* Wr = W + (size_t)colc * K;
  v8f c = {};
  for (int ks = kbeg; ks < kend; ks++) {
    int k = ks * 4 + koff;             // even -> 8B-aligned
    v2f a = *(const v2f*)(Ar + k);
    v2f b = __builtin_nontemporal_load((const v2f*)(Wr + k));
    b.x *= msk; b.y *= msk;
    c = wmma4(a, b, c);
  }
  int rbase = (lane < 16) ? 0 : 8;
#pragma unroll
  for (int r = 0; r < 8; r++)
    atomicAdd(&acc[(rbase + r) * 112 + col], c[r]);
}

__global__ void k_fc1post(const float* __restrict__ acc, const float* __restrict__ b1,
                          float* __restrict__ a1, int Bn, int F1) {
  int i = blockIdx.x * blockDim.x + threadIdx.x;
  if (i >= Bn * F1) return;
  int b = i / F1, j = i % F1;
  a1[i] = fmaxf(acc[b * 112 + j] + b1[j], 0.0f);
}

// ---------------------------------------------------------------- FC2+FC3
__global__ void k_fc23(const float* __restrict__ a1,
                       const float* __restrict__ W2, const float* __restrict__ b2,
                       const float* __restrict__ W3, const float* __restrict__ b3,
                       float* __restrict__ out, int Bn, int F1, int F2) {
  __shared__ float a2[16 * 16];
  int t = threadIdx.x;
  if (t < Bn * F2) {
    int b = t / F2, j = t % F2;
    float s = b2[j];
    for (int k = 0; k < F1; k++) s += a1[b * F1 + k] * W2[j * F1 + k];
    a2[b * 16 + j] = fmaxf(s, 0.0f);
  }
  __syncthreads();
  if (t < Bn) {
    float s = b3[0];
    for (int k = 0; k < F2; k++) s += a2[t * 16 + k] * W3[k];
    out[t] = s;
  }
}

// ================================================================ launch
extern "C" void kernel_launch(void* const* d_in, const int* in_sizes, int n_in,
                              void* d_out, int out_size, void* d_ws, size_t ws_size,
                              hipStream_t stream) {
  (void)n_in; (void)out_size; (void)ws_size;
  const float*     x     = (const float*)d_in[0];
  const long long* eidx  = (const long long*)d_in[1];
  const float*     eattr = (const float*)d_in[2];
  const float* W1l = (const float*)d_in[4],  *b1l = (const float*)d_in[5];
  const float* W1r = (const float*)d_in[6],  *b1r = (const float*)d_in[7];
  const float* We1 = (const float*)d_in[8],  *att1 = (const float*)d_in[9];
  const float* bias1 = (const float*)d_in[10];
  const float* W2l = (const float*)d_in[11], *b2l = (const float*)d_in[12];
  const float* W2r = (const float*)d_in[13], *b2r = (const float*)d_in[14];
  const float* We2 = (const float*)d_in[15], *att2 = (const float*)d_in[16];
  const float* bias2 = (const float*)d_in[17];
  const float* fcW1 = (const float*)d_in[18], *fcb1 = (const float*)d_in[19];
  const float* fcW2 = (const float*)d_in[20], *fcb2 = (const float*)d_in[21];
  const float* fcW3 = (const float*)d_in[22], *fcb3 = (const float*)d_in[23];

  int Fn = in_sizes[4] / 9;                  // 64
  int N  = in_sizes[0] / Fn;                 // 80000
  int E  = in_sizes[2] / 16;                 // 2.56M (Fe=16)
  int EP = E + N;
  int F1 = in_sizes[19];                     // 100
  int F2 = in_sizes[21];                     // 10
  int NN = in_sizes[18] / (F1 * 20);         // num_nodes = 5000
  int Bn = N / NN;                           // 16
  int K  = NN * 20;                          // 100000

  const long long* src = eidx;
  const long long* dst = eidx + E;

  // Workspace carve (~84 MB); node-sized state is L2-resident on MI455X.
  char* w = (char*)d_ws;
  auto carve = [&](size_t nfloats) -> float* {
    float* p = (float*)w;
    w += (nfloats * sizeof(float) + 255) & ~(size_t)255;
    return p;
  };
  float* cnt    = carve(N);
  float* lattr  = carve((size_t)N * 16);
  float* xl1    = carve((size_t)N * 16);
  float* xr1    = carve((size_t)N * 16);
  float* h1     = carve((size_t)N * 16);
  float* xl2    = carve((size_t)N * 20);
  float* xr2    = carve((size_t)N * 20);
  float* h2     = carve((size_t)N * 20);
  float* logitb = carve((size_t)EP * 4);     // reused in place for exp
  float* lmax   = carve((size_t)N * 4);
  float* denom  = carve((size_t)N * 4);
  float* fcacc  = carve(16 * 112);
  float* a1     = carve((size_t)16 * F1);

  dim3 b256(256);
  int gE  = (E + 255) / 256;
  int gEP = (EP + 255) / 256;
  int tilesN = (N + 15) / 16;
  int gTile = (tilesN + 7) / 8;              // 8 waves / block

  // ---- self-loop edge attrs (fill_value='mean')
  hipMemsetAsync(cnt, 0, (size_t)N * 4, stream);
  hipMemsetAsync(lattr, 0, (size_t)N * 16 * 4, stream);
  k_loop_accum<<<gE, b256, 0, stream>>>(dst, eattr, cnt, lattr, E);
  k_loop_div<<<(N * 16 + 255) / 256, b256, 0, stream>>>(lattr, cnt, N);

  // ---- layer 1
  k_node_lin1<<<gTile, b256, 0, stream>>>(x, W1l, b1l, W1r, b1r, xl1, xr1, N, Fn, 9);
  k_init_bias<<<(N * 16 + 255) / 256, b256, 0, stream>>>(h1, bias1, N, 16, 9);
  k_init_neginf<<<(N * 3 + 255) / 256, b256, 0, stream>>>(lmax, N * 3);
  hipMemsetAsync(denom, 0, (size_t)N * 3 * 4, stream);
  k_edge_logits<3, 3, 16><<<gEP, b256, 0, stream>>>(src, dst, eattr, lattr, We1, att1,
                                                    xl1, xr1, logitb, lmax, E, EP);
  k_edge_exp<3><<<gEP, b256, 0, stream>>>(dst, logitb, lmax, denom, E, EP);
  k_edge_scatter<3, 3, 16, 16><<<gEP, b256, 0, stream>>>(src, dst, logitb, denom,
                                                         xl1, h1, E, EP);
  k_relu<<<(N * 16 + 255) / 256, b256, 0, stream>>>(h1, N * 16);

  // ---- layer 2
  k_node_lin2<<<gTile, b256, 0, stream>>>(h1, W2l, b2l, W2r, b2r, xl2, xr2, N);
  k_init_bias<<<(N * 20 + 255) / 256, b256, 0, stream>>>(h2, bias2, N, 20, 20);
  k_init_neginf<<<(N * 4 + 255) / 256, b256, 0, stream>>>(lmax, N * 4);
  hipMemsetAsync(denom, 0, (size_t)N * 4 * 4, stream);
  k_edge_logits<4, 5, 20><<<gEP, b256, 0, stream>>>(src, dst, eattr, lattr, We2, att2,
                                                    xl2, xr2, logitb, lmax, E, EP);
  k_edge_exp<4><<<gEP, b256, 0, stream>>>(dst, logitb, lmax, denom, E, EP);
  k_edge_scatter<4, 5, 20, 20><<<gEP, b256, 0, stream>>>(src, dst, logitb, denom,
                                                         xl2, h2, E, EP);
  k_relu<<<(N * 20 + 255) / 256, b256, 0, stream>>>(h2, N * 20);

  // ---- MLP head: FC1 via WMMA (M=16 == batch), K split across 256 wave-slices
  hipMemsetAsync(fcacc, 0, 16 * 112 * 4, stream);
  int ntiles = (F1 + 15) / 16;               // 7
  int slices = 256;
  dim3 g1(ntiles, slices / 8);
  k_fc1_wmma<<<g1, b256, 0, stream>>>(h2, fcW1, fcacc, Bn, K, F1, slices);
  k_fc1post<<<(Bn * F1 + 255) / 256, b256, 0, stream>>>(fcacc, fcb1, a1, Bn, F1);
  k_fc23<<<1, 256, 0, stream>>>(a1, fcW2, fcb2, fcW3, fcb3, (float*)d_out, Bn, F1, F2);
}